// MultiheadAttention_47098611368038
// MI455X (gfx1250) — compile-verified
//
#include <hip/hip_runtime.h>
#include <hip/hip_bf16.h>
#include <math.h>

#define D_MODEL 1024
#define N_HEAD  16
#define D_HEAD  64
#define N_TOKEN 77
#define N_REF   8
#define SEQ2    693        // N_TOKEN * (1 + N_REF)
#define WSZ     85         // N_TOKEN + N_REF
#define BATCH   64
#define ALPHA_C 0.5f
#define EPS_C   1e-12f
#define TPAD    704        // 44 * 16 (693 padded to multiple of 16)

typedef __attribute__((ext_vector_type(2))) float        v2f;
typedef __attribute__((ext_vector_type(8))) float        v8f;
typedef __attribute__((ext_vector_type(4))) unsigned int v4u;
typedef __attribute__((ext_vector_type(8))) int          v8i;
typedef __attribute__((ext_vector_type(4))) int          v4i;

#if __has_builtin(__builtin_amdgcn_tensor_load_to_lds) && \
    __has_builtin(__builtin_amdgcn_s_wait_tensorcnt)
#define USE_TDM 1
#else
#define USE_TDM 0
#endif

__device__ __forceinline__ v8f wmma_f32x4(v2f a, v2f b, v8f c) {
    // D = A(16x4) * B(4x16) + C(16x16), all fp32 — v_wmma_f32_16x16x4_f32
    return __builtin_amdgcn_wmma_f32_16x16x4_f32(
        /*neg_a=*/false, a, /*neg_b=*/false, b,
        /*c_mod=*/(short)0, c, /*reuse_a=*/false, /*reuse_b=*/false);
}

// ---------------------------------------------------------------------------
// C[M,1024] = X[M,1024] @ W[1024,1024]^T + bias
// Block: 256 threads (8 waves). Block tile: 32 rows x 256 cols.
// Wave (wy in 0..1, wx in 0..3) computes a 16x64 tile: A-frag reused across
// 4 accumulators -> 1.25 b64 loads per WMMA.
// Grid: x = column block (4), y = row block (M/32). x iterates fastest so a
// row tile of X stays L2-hot across its column blocks; X streams from HBM
// exactly once and W (4MB) is L2-resident.
// HEAD_MAJOR != 0: scatter output as [b, h, t, 64] with b = m / SEQLEN
// (SEQLEN is a template constant so the div lowers to magic multiplies).
// ---------------------------------------------------------------------------
template <int SEQLEN, int HEAD_MAJOR>
__global__ __launch_bounds__(256)
void proj_gemm(const float* __restrict__ X, const float* __restrict__ W,
               const float* __restrict__ bias, float* __restrict__ out) {
    const int lane = threadIdx.x & 31;
    const int wave = threadIdx.x >> 5;
    const int wy   = wave & 1;
    const int wx   = wave >> 1;
    const int l16  = lane & 15;
    const int hi   = lane >> 4;          // 0: lanes 0-15, 1: lanes 16-31
    const int koff = hi << 1;            // K sub-offset 0 or 2

    const int m0   = blockIdx.y * 32 + wy * 16;
    const int n0   = blockIdx.x * 256 + wx * 64;
    const int arow = m0 + l16;

    const float* Xr  = X + (size_t)arow * D_MODEL;
    const float* Wr0 = W + (size_t)(n0 +  0 + l16) * D_MODEL;  // B[k][n] = W[n][k]
    const float* Wr1 = W + (size_t)(n0 + 16 + l16) * D_MODEL;
    const float* Wr2 = W + (size_t)(n0 + 32 + l16) * D_MODEL;
    const float* Wr3 = W + (size_t)(n0 + 48 + l16) * D_MODEL;

    v8f acc[4] = {};
    for (int kk = 0; kk < D_MODEL; kk += 4) {
        v2f a  = *(const v2f*)(Xr  + kk + koff);   // (X[row][k], X[row][k+1])
        v2f b0 = *(const v2f*)(Wr0 + kk + koff);
        v2f b1 = *(const v2f*)(Wr1 + kk + koff);
        v2f b2 = *(const v2f*)(Wr2 + kk + koff);
        v2f b3 = *(const v2f*)(Wr3 + kk + koff);
        acc[0] = wmma_f32x4(a, b0, acc[0]);
        acc[1] = wmma_f32x4(a, b1, acc[1]);
        acc[2] = wmma_f32x4(a, b2, acc[2]);
        acc[3] = wmma_f32x4(a, b3, acc[3]);
    }

    #pragma unroll
    for (int c = 0; c < 4; ++c) {
        const int   ncol = n0 + c * 16 + l16;
        const float bvv  = bias[ncol];
        if (!HEAD_MAJOR) {
            #pragma unroll
            for (int j = 0; j < 8; ++j) {
                int m = m0 + j + (hi << 3);      // VGPR j: row j (lo) / j+8 (hi)
                out[(size_t)m * D_MODEL + ncol] = acc[c][j] + bvv;
            }
        } else {
            const int h = ncol >> 6, d = ncol & 63;
            #pragma unroll
            for (int j = 0; j < 8; ++j) {
                int m  = m0 + j + (hi << 3);
                int bb = m / SEQLEN, t = m - bb * SEQLEN;
                out[(((size_t)bb * N_HEAD + h) * SEQLEN + t) * D_HEAD + d] =
                    acc[c][j] + bvv;
            }
        }
    }
}

// ---------------------------------------------------------------------------
// Attention core: one block per (batch, head, 16-row s-tile). 256 threads.
// Phase 0: TDM async-stages the 16x693 mask tile into LDS (overlaps Phase 1).
// Phase 1: raw scores (16 x 693, padded 704) into LDS via WMMA (8 waves).
// Phase 1b: apply mask from LDS (TDM zero-fills rows past the tensor edge,
//           which also zeroes the clamped rows of the partial last s-tile).
// Phase 2: target softmax + 8 ref softmaxes + weighting + normalization.
// Phase 3: out tile (16 x 64) = w @ Vh via WMMA (4 waves), scatter to ctx.
// ---------------------------------------------------------------------------
__global__ __launch_bounds__(256)
void attn_core(const float* __restrict__ qh, const float* __restrict__ kh,
               const float* __restrict__ vh, const float* __restrict__ mask,
               const float* __restrict__ weight, float* __restrict__ ctx) {
    __shared__ float sc[16][TPAD];
    __shared__ float lmask[16][SEQ2];

    const int blk    = blockIdx.x;
    const int s_tile = blk % 5;
    const int h      = (blk / 5) % N_HEAD;
    const int b      = blk / (5 * N_HEAD);
    const int s0     = s_tile * 16;

    const int lane = threadIdx.x & 31;
    const int wave = threadIdx.x >> 5;
    const int l16  = lane & 15;
    const int hi   = lane >> 4;
    const int koff = hi << 1;
    const int tid  = threadIdx.x;

#if USE_TDM
    // ---- Phase 0: TDM DMA of the mask tile into LDS --------------------------
    if (wave == 0) {
        const unsigned long long gaddr =
            (unsigned long long)(uintptr_t)(mask + ((size_t)b * N_TOKEN + s0) * SEQ2);
        const unsigned int lds_base = (unsigned int)(uintptr_t)&lmask[0][0];
        const int rows_rem = N_TOKEN - s0;   // tensor extent from tile start

        v4u g0;
        g0.x = 1u;                                   // count=1, user descriptor
        g0.y = lds_base;                             // LDS byte address
        g0.z = (unsigned int)gaddr;                  // global_addr[31:0]
        g0.w = (unsigned int)((gaddr >> 32) & 0x1FFFFFFull) | (2u << 30); // type=2

        v8i g1;
        g1[0] = (2 << 16);                 // data_size = 4 bytes; no flags
        g1[1] = (SEQ2 & 0xFFFF) << 16;     // tensor_dim0 lo16 (693)
        g1[2] = (rows_rem & 0xFFFF) << 16; // dim0 hi16 = 0 | tensor_dim1 lo16
        g1[3] = (SEQ2 & 0xFFFF) << 16;     // dim1 hi16 = 0 | tile_dim0 = 693
        g1[4] = 16;                        // tile_dim1 = 16, tile_dim2 = 0
        g1[5] = SEQ2;                      // tensor_dim0_stride lo32 = 693
        g1[6] = 0;                         // stride0 hi16 | tensor_dim1_stride lo16
        g1[7] = 0;                         // tensor_dim1_stride hi32
        v4i gz4 = {0, 0, 0, 0};
        v8i gz8 = {0, 0, 0, 0, 0, 0, 0, 0};

        // 6-arg toolchain variant: (g0, g1, g2, g3, g4, cpol)
        __builtin_amdgcn_tensor_load_to_lds(g0, g1, gz4, gz4, gz8, 0);
    }
#endif

    const float* qbase = qh + ((size_t)b * N_HEAD + h) * N_TOKEN * D_HEAD;
    const float* kbase = kh + ((size_t)b * N_HEAD + h) * SEQ2 * D_HEAD;
    const float* vbase = vh + ((size_t)b * N_HEAD + h) * SEQ2 * D_HEAD;

    // ---- Phase 1: raw scores = Q K^T / 8 (mask applied later from LDS) -------
    {
        int qrow = s0 + l16;
        if (qrow > N_TOKEN - 1) qrow = N_TOKEN - 1;   // clamp partial tile
        const float* qr = qbase + (size_t)qrow * D_HEAD;
        v2f afrag[16];
        #pragma unroll
        for (int kd = 0; kd < 16; ++kd)
            afrag[kd] = *(const v2f*)(qr + kd * 4 + koff);

        for (int tt = wave; tt < 44; tt += 8) {
            const int t0   = tt * 16;
            const int tcol = t0 + l16;
            const int tcl  = tcol < SEQ2 ? tcol : SEQ2 - 1;
            const float* kr = kbase + (size_t)tcl * D_HEAD;
            v8f acc = {};
            #pragma unroll
            for (int kd = 0; kd < 16; ++kd) {
                v2f bfrag = *(const v2f*)(kr + kd * 4 + koff);  // Kh^T tile
                acc = wmma_f32x4(afrag[kd], bfrag, acc);
            }
            #pragma unroll
            for (int j = 0; j < 8; ++j) {
                const int r = j + (hi << 3);
                sc[r][tcol] = (tcol < SEQ2) ? acc[j] * 0.125f : 0.0f; // /sqrt(64)
            }
        }
    }
#if USE_TDM
    __builtin_amdgcn_s_wait_tensorcnt(0);   // wave 0: DMA done; others: no-op
#endif
    __syncthreads();

    // ---- Phase 1b: apply mask (zero rows past the tensor edge) ---------------
    #pragma unroll
    for (int r = 0; r < 16; ++r) {
        for (int t = tid; t < SEQ2; t += 256) {
#if USE_TDM
            sc[r][t] *= lmask[r][t];
#else
            const int srow = s0 + r;
            sc[r][t] = (srow < N_TOKEN)
                     ? sc[r][t] * mask[((size_t)b * N_TOKEN + srow) * SEQ2 + t]
                     : 0.0f;
#endif
        }
    }
    __syncthreads();

    // ---- Phase 2: softmaxes + weighting --------------------------------------
    if (tid < 144) {                       // 16 rows x 9 segments of length 77
        const int row = tid / 9;
        const int seg = tid % 9;
        const int c0  = (seg == 0) ? 0 : N_TOKEN + (seg - 1) * N_TOKEN;
        float mx = -3.0e38f;
        for (int j = 0; j < N_TOKEN; ++j) mx = fmaxf(mx, sc[row][c0 + j]);
        float sum = 0.0f;
        for (int j = 0; j < N_TOKEN; ++j) {
            float e = __expf(sc[row][c0 + j] - mx);
            sc[row][c0 + j] = e;
            sum += e;
        }
        if (seg == 0) {
            const float inv = (1.0f - ALPHA_C) / sum;
            for (int j = 0; j < N_TOKEN; ++j)
                sc[row][j] = sc[row][j] * inv * weight[(size_t)b * WSZ + j];
        } else {
            const float rw  = weight[(size_t)b * WSZ + N_TOKEN + (seg - 1)];
            const float inv = rw / sum;
            for (int j = 0; j < N_TOKEN; ++j) sc[row][c0 + j] *= inv;
        }
    }
    __syncthreads();
    if (tid < 16) {                        // per-row ref-norm + global norm
        const int row = tid;
        float rsum = 0.0f;
        for (int j = N_TOKEN; j < SEQ2; ++j) rsum += sc[row][j];
        const float sref = ALPHA_C / (rsum + EPS_C);
        float tsum = 0.0f;
        for (int j = 0; j < N_TOKEN; ++j) tsum += sc[row][j];
        const float total  = tsum + rsum * sref;
        const float invtot = 1.0f / (total + EPS_C);
        for (int j = 0; j < N_TOKEN; ++j) sc[row][j] *= invtot;
        const float sr2 = sref * invtot;
        for (int j = N_TOKEN; j < SEQ2; ++j) sc[row][j] *= sr2;
    }
    __syncthreads();

    // ---- Phase 3: out(16x64) = w(16x693) @ Vh(693x64) ------------------------
    if (wave < 4) {
        const int d0   = wave * 16;
        const int dcol = d0 + l16;
        v8f acc = {};
        for (int tk = 0; tk < TPAD; tk += 4) {
            v2f a = *(const v2f*)(&sc[l16][tk + koff]);    // w rows from LDS
            int t0i = tk + koff;
            int t1i = t0i + 1;
            if (t0i >= SEQ2) t0i = SEQ2 - 1;               // A is zero there
            if (t1i >= SEQ2) t1i = SEQ2 - 1;
            v2f bfrag;
            bfrag.x = vbase[(size_t)t0i * D_HEAD + dcol];
            bfrag.y = vbase[(size_t)t1i * D_HEAD + dcol];
            acc = wmma_f32x4(a, bfrag, acc);
        }
        #pragma unroll
        for (int j = 0; j < 8; ++j) {
            const int r    = j + (hi << 3);
            const int srow = s0 + r;
            if (srow < N_TOKEN)
                ctx[((size_t)b * N_TOKEN + srow) * D_MODEL + h * D_HEAD + dcol] = acc[j];
        }
    }
}

// ---------------------------------------------------------------------------
extern "C" void kernel_launch(void* const* d_in, const int* in_sizes, int n_in,
                              void* d_out, int out_size, void* d_ws, size_t ws_size,
                              hipStream_t stream) {
    (void)in_sizes; (void)n_in; (void)out_size; (void)ws_size;
    const float* q      = (const float*)d_in[0];
    const float* k      = (const float*)d_in[1];
    const float* v      = (const float*)d_in[2];
    const float* mask   = (const float*)d_in[3];
    const float* weight = (const float*)d_in[4];
    const float* Wq     = (const float*)d_in[5];
    const float* bq     = (const float*)d_in[6];
    const float* Wk     = (const float*)d_in[7];
    const float* bk     = (const float*)d_in[8];
    const float* Wv     = (const float*)d_in[9];
    const float* bv     = (const float*)d_in[10];
    const float* Wp     = (const float*)d_in[11];
    const float* bp     = (const float*)d_in[12];
    float* out = (float*)d_out;

    float* ws = (float*)d_ws;
    size_t o = 0;
    float* qh  = ws + o; o += (size_t)BATCH * N_TOKEN * D_MODEL;
    float* kh  = ws + o; o += (size_t)BATCH * SEQ2   * D_MODEL;
    float* vh  = ws + o; o += (size_t)BATCH * SEQ2   * D_MODEL;
    float* ctx = ws + o;

    const dim3 blk(256);
    const dim3 gq(D_MODEL / 256, (BATCH * N_TOKEN) / 32);   // 4 x 154
    const dim3 gk(D_MODEL / 256, (BATCH * SEQ2)   / 32);    // 4 x 1386

    proj_gemm<N_TOKEN, 1><<<gq, blk, 0, stream>>>(q, Wq, bq, qh);
    proj_gemm<SEQ2,    1><<<gk, blk, 0, stream>>>(k, Wk, bk, kh);
    proj_gemm<SEQ2,    1><<<gk, blk, 0, stream>>>(v, Wv, bv, vh);

    attn_core<<<dim3(BATCH * N_HEAD * 5), blk, 0, stream>>>(qh, kh, vh, mask, weight, ctx);

    proj_gemm<N_TOKEN, 0><<<gq, blk, 0, stream>>>(ctx, Wp, bp, out);
}